// SetKernelMultiheadAttentionTokenized_55310588838261
// MI455X (gfx1250) — compile-verified
//
#include <hip/hip_runtime.h>
#include <hip/hip_bf16.h>

// ---------------------------------------------------------------------------
// SetKernelMultiheadAttentionTokenized — MI455X (gfx1250, wave32, WMMA)
//
// Pipeline: minhash sigs -> fused (jaccard -> score -> softmax -> P@V) -> proj
// All GEMMs use V_WMMA_F32_16X16X4_F32 (full fp32; problem is L2-resident and
// nowhere near compute bound, so no precision downconversion is warranted).
// sig_k tiles are staged into LDS with GLOBAL_LOAD_ASYNC_TO_LDS_B128
// (ASYNCcnt path) instead of a VGPR round trip.
// ---------------------------------------------------------------------------

typedef __attribute__((ext_vector_type(2))) float v2f;
typedef __attribute__((ext_vector_type(8))) float v8f;

constexpr int   BATCH = 4;
constexpr int   SEQL  = 1024;   // Lq == Lk
constexpr int   EMB   = 512;
constexpr int   NH    = 128;    // number of hash slots K
constexpr int   SETS  = 8;      // set size S
constexpr int   KT    = 64;     // k-tile for fused attention
constexpr long long MINHASH_PRIME = 2147483647LL;
// exponent coefficient: GAMMA * (|A|+|B|) = 0.3 * 16
constexpr float GAMMA_TOT = 4.8f;

// ---------------------------------------------------------------------------
// Kernel 1: MinHash signatures. One block per token row, 128 threads = K slots.
// ---------------------------------------------------------------------------
__global__ void minhash_kernel(const int* __restrict__ ids,  // (N, SETS)
                               const int* __restrict__ ha,   // (NH)
                               const int* __restrict__ hb,   // (NH)
                               int* __restrict__ sig)        // (N, NH)
{
    __shared__ int lid[SETS];
    const int row = blockIdx.x;
    const int k   = threadIdx.x;            // 0..127
    if (k < SETS) lid[k] = ids[row * SETS + k];
    __syncthreads();

    const long long a = (long long)ha[k];
    const long long b = (long long)hb[k];
    long long best = MINHASH_PRIME;
    #pragma unroll
    for (int s = 0; s < SETS; ++s) {
        long long h = (a * (long long)lid[s] + b) % MINHASH_PRIME;
        best = (h < best) ? h : best;
    }
    sig[row * NH + k] = (int)best;
}

// ---------------------------------------------------------------------------
// Kernel 3/5: Y(M,512) = X(M,512) @ W^T + bias, fp32 WMMA 16x16x4.
// Block: 256 threads (8 waves). Block covers 16 rows x 512 cols; each wave
// owns 64 cols (4 accumulator tiles). Fragment loads are contiguous float2:
//   A lane(l): row = r0+(l&15), ks = k0 + 2*(l>>4)      -> X[row*512 + ks]
//   B lane(l): col = n0+(l&15), ks = k0 + 2*(l>>4)      -> W[col*512 + ks]
// (W row-major (out,in); B matrix of the GEMM is W^T, i.e. B[k][n] = W[n][k].)
// ---------------------------------------------------------------------------
__global__ void gemm_xwt_bias_kernel(const float* __restrict__ X,
                                     const float* __restrict__ W,
                                     const float* __restrict__ bias,
                                     float* __restrict__ Y)
{
    const int lane = threadIdx.x & 31;
    const int wave = threadIdx.x >> 5;      // 0..7
    const int row0 = blockIdx.x * 16;
    const int col0 = wave * 64;

    const int arow = row0 + (lane & 15);
    const int koff = (lane >> 4) * 2;       // 0 or 2

    v8f zero = {0.f, 0.f, 0.f, 0.f, 0.f, 0.f, 0.f, 0.f};
    v8f acc[4] = {zero, zero, zero, zero};

    for (int k0 = 0; k0 < EMB; k0 += 4) {
        const v2f a = *(const v2f*)(X + (size_t)arow * EMB + k0 + koff);
        #pragma unroll
        for (int t = 0; t < 4; ++t) {
            const int n = col0 + t * 16 + (lane & 15);
            const v2f b = *(const v2f*)(W + (size_t)n * EMB + k0 + koff);
            acc[t] = __builtin_amdgcn_wmma_f32_16x16x4_f32(
                false, a, false, b, (short)0, acc[t], false, false);
        }
    }

    // C/D layout: VGPR r, lanes 0-15 -> M=r ; lanes 16-31 -> M=r+8 ; N = lane&15
    const int rbase = (lane >> 4) * 8;
    #pragma unroll
    for (int t = 0; t < 4; ++t) {
        const int col = col0 + t * 16 + (lane & 15);
        const float bv = bias[col];
        #pragma unroll
        for (int r = 0; r < 8; ++r) {
            Y[(size_t)(row0 + rbase + r) * EMB + col] = acc[t][r] + bv;
        }
    }
}

// ---------------------------------------------------------------------------
// Kernel 4: fused Jaccard -> score -> softmax -> P @ Vproj  (flash-style).
// Grid: (SEQL/16, BATCH). Block: 256 threads (8 waves); wave owns 64 of the
// 512 output columns. Never materializes the (B,Lq,Lk) score tensor.
// scores = exp(-4.8*(1-J)/(1+J)) <= 1, so softmax needs no max subtraction:
// P = exp(score) in [1, e]; row sums reduced deterministically (no atomics).
// sig_k tiles staged via GLOBAL_LOAD_ASYNC_TO_LDS_B128 + s_wait_asynccnt.
// ---------------------------------------------------------------------------
__global__ void attn_fused_kernel(const int*   __restrict__ sigQ,  // (B*L, NH)
                                  const int*   __restrict__ sigK,  // (B*L, NH)
                                  const float* __restrict__ Vp,    // (B*L, EMB)
                                  float*       __restrict__ Ctx)   // (B*L, EMB)
{
    constexpr int SKP = 132;                 // padded sigK row (ints, 16B mult)
    constexpr int LPP = 66;                  // padded P row (floats, 8B align)
    __shared__ __align__(16) int   lsq[16 * NH];
    __shared__ __align__(16) int   lsk[KT * SKP];
    __shared__ __align__(8)  float lp [16 * LPP];
    __shared__ float lrow[16];

    const int b   = blockIdx.y;
    const int q0  = blockIdx.x * 16;
    const int tid = threadIdx.x;
    const int lane = tid & 31;
    const int wave = tid >> 5;
    const int col0 = wave * 64;

    // stage sig_q for the 16 query rows (2048 ints)
    for (int i = tid; i < 16 * NH; i += 256) {
        const int r = i / NH, c = i % NH;
        lsq[i] = sigQ[(size_t)(b * SEQL + q0 + r) * NH + c];
    }
    if (tid < 16) lrow[tid] = 0.f;

    v8f zero = {0.f, 0.f, 0.f, 0.f, 0.f, 0.f, 0.f, 0.f};
    v8f acc[4] = {zero, zero, zero, zero};

    const int arow = lane & 15;
    const int koff = (lane >> 4) * 2;        // 0 or 2
    const int myq  = tid >> 4;               // 0..15 (query row this thread scores)
    const int mykk = (tid & 15) * 4;         // 4 consecutive keys per thread

    // per-thread fixed slice for the async sig_k staging:
    // KT rows * (NH/4) 16B-chunks = 2048 chunks, 8 per thread, no divergence.
    const int nchunk = NH / 4;               // 32 chunks per row

    for (int kt = 0; kt < SEQL; kt += KT) {
        __syncthreads();                     // prior readers of lsk/lp done

        // ---- async global -> LDS staging of the sig_k tile (ASYNCcnt path)
        #pragma unroll
        for (int ii = 0; ii < 8; ++ii) {
            const int i  = tid + ii * 256;
            const int r  = i / nchunk;       // tile row 0..63
            const int c4 = (i % nchunk) * 4; // int index within row
            const unsigned lds_off =
                (unsigned)(uintptr_t)(const void*)&lsk[r * SKP + c4];
            const unsigned long long gaddr = (unsigned long long)(uintptr_t)
                (sigK + (size_t)(b * SEQL + kt + r) * NH + c4);
            asm volatile("global_load_async_to_lds_b128 %0, %1, off"
                         :: "v"(lds_off), "v"(gaddr)
                         : "memory");
        }
        // prefetch next tile toward the WGP while we score this one
        if (kt + KT < SEQL) {
            __builtin_prefetch(sigK + (size_t)(b * SEQL + kt + KT) * NH +
                               (tid & 63) * 2 * NH, 0, 3);
        }
        asm volatile("s_wait_asynccnt 0x0" ::: "memory");
        __syncthreads();                     // tile visible to all waves

        // P tile: 16 x KT entries, 4 per thread; 128 compares each (int4 LDS)
        #pragma unroll
        for (int j = 0; j < 4; ++j) {
            const int kk = mykk + j;
            const int4* sq4 = (const int4*)&lsq[myq * NH];
            const int4* sk4 = (const int4*)&lsk[kk * SKP];
            int cnt = 0;
            #pragma unroll
            for (int i = 0; i < NH / 4; ++i) {
                const int4 a4 = sq4[i];
                const int4 b4 = sk4[i];
                cnt += (a4.x == b4.x) + (a4.y == b4.y) +
                       (a4.z == b4.z) + (a4.w == b4.w);
            }
            const float J     = (float)cnt * (1.0f / (float)NH);
            const float score = __expf(-GAMMA_TOT * (1.0f - J) / (1.0f + J));
            lp[myq * LPP + kk] = __expf(score);   // softmax numerator (<= e)
        }
        __syncthreads();

        // deterministic row-sum reduction (fixed order, no float atomics)
        if (tid < 16) {
            float s = 0.f;
            for (int kk = 0; kk < KT; ++kk) s += lp[tid * LPP + kk];
            lrow[tid] += s;
        }

        // acc += P(16 x KT) @ Vp(KT x 64cols) via fp32 WMMA, K-step = 4
        for (int ks = 0; ks < KT; ks += 4) {
            const v2f a = *(const v2f*)&lp[arow * LPP + ks + koff];
            const float* vrow = Vp + (size_t)(b * SEQL + kt + ks + koff) * EMB;
            #pragma unroll
            for (int t = 0; t < 4; ++t) {
                const int n = col0 + t * 16 + (lane & 15);
                v2f bb;
                bb.x = vrow[n];              // B[ks+koff  ][n]
                bb.y = vrow[EMB + n];        // B[ks+koff+1][n]
                acc[t] = __builtin_amdgcn_wmma_f32_16x16x4_f32(
                    false, a, false, bb, (short)0, acc[t], false, false);
            }
        }
    }
    __syncthreads();                         // lrow complete

    const int rbase = (lane >> 4) * 8;
    #pragma unroll
    for (int t = 0; t < 4; ++t) {
        const int col = col0 + t * 16 + (lane & 15);
        #pragma unroll
        for (int r = 0; r < 8; ++r) {
            const float denom = lrow[rbase + r];
            Ctx[(size_t)(b * SEQL + q0 + rbase + r) * EMB + col] =
                acc[t][r] / denom;
        }
    }
}

// ---------------------------------------------------------------------------
// Launch. Inputs (setup_inputs order):
//  0 query 1 key 2 value 3 token_sets_q 4 token_sets_k 5 hash_a 6 hash_b
//  7 Wq 8 bq 9 Wk 10 bk 11 Wv 12 bv 13 Wo 14 bo
// q/k projections are dead code in the reference -> skipped entirely.
// ---------------------------------------------------------------------------
extern "C" void kernel_launch(void* const* d_in, const int* in_sizes, int n_in,
                              void* d_out, int out_size, void* d_ws, size_t ws_size,
                              hipStream_t stream) {
    (void)in_sizes; (void)n_in; (void)out_size; (void)ws_size;

    const float* value = (const float*)d_in[2];
    const int*   tsq   = (const int*)d_in[3];
    const int*   tsk   = (const int*)d_in[4];
    const int*   ha    = (const int*)d_in[5];
    const int*   hb    = (const int*)d_in[6];
    const float* Wv    = (const float*)d_in[11];
    const float* bv    = (const float*)d_in[12];
    const float* Wo    = (const float*)d_in[13];
    const float* bo    = (const float*)d_in[14];
    float*       out   = (float*)d_out;

    // workspace layout (~20 MB): sig_q | sig_k | v_proj | ctx
    const size_t nrows = (size_t)BATCH * SEQL;
    int*   sigq  = (int*)d_ws;
    int*   sigk  = sigq + nrows * NH;
    float* vproj = (float*)(sigk + nrows * NH);
    float* ctx   = vproj + nrows * EMB;

    // 1-2: MinHash signatures for both sides
    minhash_kernel<<<(int)nrows, NH, 0, stream>>>(tsq, ha, hb, sigq);
    minhash_kernel<<<(int)nrows, NH, 0, stream>>>(tsk, ha, hb, sigk);

    // 3: v = value @ Wv^T + bv   (fp32 WMMA)
    gemm_xwt_bias_kernel<<<(int)(nrows / 16), 256, 0, stream>>>(value, Wv, bv, vproj);

    // 4: fused jaccard -> softmax -> ctx = attn @ v   (fp32 WMMA, async LDS)
    attn_fused_kernel<<<dim3(SEQL / 16, BATCH), 256, 0, stream>>>(sigq, sigk, vproj, ctx);

    // 5: out = ctx @ Wo^T + bo   (fp32 WMMA)
    gemm_xwt_bias_kernel<<<(int)(nrows / 16), 256, 0, stream>>>(ctx, Wo, bo, out);
}